// FASASaliencyDetector_37563783971173
// MI455X (gfx1250) — compile-verified
//
#include <hip/hip_runtime.h>
#include <math.h>

typedef __attribute__((ext_vector_type(2))) float v2f;
typedef __attribute__((ext_vector_type(8))) float v8f;

#define HH 4096
#define WW 4096
#define HWPIX (HH * WW)
#define NBIN 512
#define NB2 512          // blocks for binning pass

// ---- workspace layout (bytes) ----
#define OFF_MM        0            // int[8]: cmin[3], cmax[3]
#define OFF_BIN       1024         // float[2560] binstats
#define OFF_CONTRAST  16384        // float[512]
#define OFF_NORMSAL   20480        // float[512]
#define OFF_EMAT      32768        // float[512*512] = 1 MB
#define OFF_PART      2097152      // float[512*2560] = 5.24 MB
#define OFF_LIN       8388608      // u16[HWPIX]     = 33.55 MB
#define OFF_TMP       41943040     // float[HWPIX]   = 67.1 MB

__constant__ float c_mean[4] = {0.5555f, 0.6449f, 0.0002f, 0.0063f};
__constant__ float c_covinv[16] = {
    43.3777f, 1.7633f, -0.4059f, 1.0997f,
    1.7633f, 40.7221f, -0.0165f, 0.0447f,
    -0.4059f, -0.0165f, 87.0455f, -3.2744f,
    1.0997f, 0.0447f, -3.2744f, 125.1503f};

__device__ __forceinline__ int refl101(int p, int n) {
  if (p < 0) p = -p;
  if (p >= n) p = 2 * n - 2 - p;
  return p;
}

// ---------------- K0: init min/max ----------------
__global__ void init_kernel(int* mm) {
  int t = threadIdx.x;
  if (t < 3) { mm[t] = 1 << 30; mm[3 + t] = -(1 << 30); }
}

// ---------------- K1: per-channel min/max of floor(img*255) ----------------
__global__ __launch_bounds__(256) void minmax_kernel(const float* __restrict__ img, int* mm) {
  __shared__ int smn[256], smx[256];
  int lmn[3] = {1 << 30, 1 << 30, 1 << 30};
  int lmx[3] = {-(1 << 30), -(1 << 30), -(1 << 30)};
  for (long i = (long)blockIdx.x * 256 + threadIdx.x; i < HWPIX; i += (long)gridDim.x * 256) {
#pragma unroll
    for (int c = 0; c < 3; ++c) {
      int v = (int)floorf(img[(size_t)c * HWPIX + i] * 255.0f);
      lmn[c] = min(lmn[c], v);
      lmx[c] = max(lmx[c], v);
    }
  }
  for (int c = 0; c < 3; ++c) {
    smn[threadIdx.x] = lmn[c];
    smx[threadIdx.x] = lmx[c];
    __syncthreads();
    for (int s = 128; s > 0; s >>= 1) {
      if (threadIdx.x < s) {
        smn[threadIdx.x] = min(smn[threadIdx.x], smn[threadIdx.x + s]);
        smx[threadIdx.x] = max(smx[threadIdx.x], smx[threadIdx.x + s]);
      }
      __syncthreads();
    }
    if (threadIdx.x == 0) { atomicMin(&mm[c], smn[0]); atomicMax(&mm[3 + c], smx[0]); }
    __syncthreads();
  }
}

// ---------------- K2: quantize + per-block bin stats + store bin ids ----------------
__global__ __launch_bounds__(256) void bin_kernel(const float* __restrict__ img,
                                                  const int* __restrict__ mm,
                                                  float* __restrict__ partials,
                                                  unsigned short* __restrict__ lin16) {
  __shared__ float acc[NBIN * 5];
  for (int j = threadIdx.x; j < NBIN * 5; j += 256) acc[j] = 0.0f;
  float cmn[3], stp[3];
#pragma unroll
  for (int c = 0; c < 3; ++c) {
    cmn[c] = (float)mm[c];
    stp[c] = fmaxf(((float)mm[3 + c] - cmn[c]) / 8.0f, 1e-12f);
  }
  __syncthreads();
  for (long i = (long)blockIdx.x * 256 + threadIdx.x; i < HWPIX; i += (long)NB2 * 256) {
    int y = (int)(i >> 12);
    int x = (int)(i & 4095);
    int q[3];
#pragma unroll
    for (int c = 0; c < 3; ++c) {
      float lab = floorf(img[(size_t)c * HWPIX + i] * 255.0f);
      int qq = (int)floorf((lab - cmn[c]) / stp[c]);
      q[c] = min(max(qq, 0), 7);
    }
    int lin = (q[0] * 8 + q[1]) * 8 + q[2];
    lin16[i] = (unsigned short)lin;
    float xf = (float)x, yf = (float)y;
    atomicAdd(&acc[lin * 5 + 0], 1.0f);
    atomicAdd(&acc[lin * 5 + 1], xf);
    atomicAdd(&acc[lin * 5 + 2], yf);
    atomicAdd(&acc[lin * 5 + 3], xf * xf);
    atomicAdd(&acc[lin * 5 + 4], yf * yf);
  }
  __syncthreads();
  for (int j = threadIdx.x; j < NBIN * 5; j += 256)
    partials[(size_t)blockIdx.x * (NBIN * 5) + j] = acc[j];
}

// ---------------- K3: deterministic fixed-order reduce of partials ----------------
__global__ __launch_bounds__(256) void reduce_bins(const float* __restrict__ partials,
                                                   float* __restrict__ binstats) {
  int j = blockIdx.x * 256 + threadIdx.x;
  if (j < NBIN * 5) {
    float s = 0.0f;
    for (int b = 0; b < NB2; ++b) s += partials[(size_t)b * (NBIN * 5) + j];
    binstats[j] = s;
  }
}

// ---------------- K4a: E matrix (512x512) + contrast = D @ hist ----------------
__global__ __launch_bounds__(256) void emat_kernel(const int* __restrict__ mm,
                                                   const float* __restrict__ binstats,
                                                   float* __restrict__ Emat,
                                                   float* __restrict__ contrast) {
  __shared__ float red[256];
  int m = blockIdx.x;
  float cmn[3], stp[3];
#pragma unroll
  for (int c = 0; c < 3; ++c) {
    cmn[c] = (float)mm[c];
    stp[c] = fmaxf(((float)mm[3 + c] - cmn[c]) / 8.0f, 1e-12f);
  }
  int q0 = m >> 6, q1 = (m >> 3) & 7, q2 = m & 7;
  float cm0 = cmn[0] + q0 * stp[0];
  float cm1 = cmn[1] + q1 * stp[1];
  float cm2 = cmn[2] + q2 * stp[2];
  float lsum = 0.0f;
  for (int k = threadIdx.x; k < NBIN; k += 256) {
    int k0 = k >> 6, k1 = (k >> 3) & 7, k2 = k & 7;
    float d0 = cm0 - (cmn[0] + k0 * stp[0]);
    float d1 = cm1 - (cmn[1] + k1 * stp[1]);
    float d2c = cm2 - (cmn[2] + k2 * stp[2]);
    float d2 = d0 * d0 + d1 * d1 + d2c * d2c;
    Emat[(size_t)m * NBIN + k] = expf(-d2 * (1.0f / 512.0f));  // 2*sigma_c^2 = 512
    lsum += sqrtf(d2) * binstats[k * 5 + 0];
  }
  red[threadIdx.x] = lsum;
  __syncthreads();
  for (int s = 128; s > 0; s >>= 1) {
    if (threadIdx.x < s) red[threadIdx.x] += red[threadIdx.x + s];
    __syncthreads();
  }
  if (threadIdx.x == 0) contrast[m] = red[0];
}

// ---------------- K4b: bilateral GEMMs via V_WMMA_F32_16X16X4_F32 ----------------
// Single block, 512 threads = 16 waves. Two GEMM passes of (512x512)x(512x16).
__global__ __launch_bounds__(512) void bilateral_wmma_kernel(const float* __restrict__ Emat,
                                                             const float* __restrict__ binstats,
                                                             const float* __restrict__ contrast,
                                                             float* __restrict__ normSal) {
  __shared__ float Bm[NBIN * 16];    // B operand, row-major [k][n]   (32 KB)
  __shared__ float outA[NBIN * 8];   // GEMM results, cols 0..6 used  (16 KB)
  __shared__ float s_mask[NBIN];     // occupied-bin mask
  __shared__ float s_num[NBIN];      // E @ (sal*mask)
  __shared__ float rmin[NBIN], rmax[NBIN];

  int tid = threadIdx.x;
  // stage B: cols = hist, cx, cy, cx2, cy2, maskf, 0...
  {
    int m = tid;  // 512 threads
    float h = binstats[m * 5 + 0];
    Bm[m * 16 + 0] = h;
    Bm[m * 16 + 1] = binstats[m * 5 + 1];
    Bm[m * 16 + 2] = binstats[m * 5 + 2];
    Bm[m * 16 + 3] = binstats[m * 5 + 3];
    Bm[m * 16 + 4] = binstats[m * 5 + 4];
    Bm[m * 16 + 5] = (h > 0.0f) ? 1.0f : 0.0f;
#pragma unroll
    for (int c = 6; c < 16; ++c) Bm[m * 16 + c] = 0.0f;
  }
  __syncthreads();

  int wave = tid >> 5;
  int lane = tid & 31;
  int laneM = lane & 15;   // M (for A) or N (for B/C)
  int laneHi = lane >> 4;  // 0 or 1

  // -------- pass 1: [hist,cx,cy,cx2,cy2,maskf] --------
  for (int t = 0; t < 2; ++t) {
    int m0 = (wave * 2 + t) * 16;
    v8f c = {};
    for (int k0 = 0; k0 < NBIN; k0 += 4) {
      int kb = k0 + laneHi * 2;
      v2f a, b;
      a.x = Emat[(size_t)(m0 + laneM) * NBIN + kb];
      a.y = Emat[(size_t)(m0 + laneM) * NBIN + kb + 1];
      b.x = Bm[kb * 16 + laneM];
      b.y = Bm[(kb + 1) * 16 + laneM];
      c = __builtin_amdgcn_wmma_f32_16x16x4_f32(false, a, false, b, (short)0, c, false, false);
    }
    if (laneM < 6) {
#pragma unroll
      for (int r = 0; r < 8; ++r) {
        int row = m0 + r + laneHi * 8;
        outA[row * 8 + laneM] = c[r];
      }
    }
  }
  __syncthreads();

  // -------- per-bin math: shape prob, saliency; rebuild B with sal*mask --------
  {
    int m = tid;
    float h = Bm[m * 16 + 0];
    float maskf = (h > 0.0f) ? 1.0f : 0.0f;
    float nrm = fmaxf(outA[m * 8 + 0], 1e-8f);
    float mx = outA[m * 8 + 1] / nrm;
    float my = outA[m * 8 + 2] / nrm;
    float mx2 = outA[m * 8 + 3] / nrm;
    float my2 = outA[m * 8 + 4] / nrm;
    float vx = fmaxf(mx2 - mx * mx, 0.0f);
    float vy = fmaxf(my2 - my * my, 0.0f);
    float g[4];
    g[0] = sqrtf(12.0f * vx) / (float)WW;
    g[1] = sqrtf(12.0f * vy) / (float)HH;
    g[2] = (mx - (float)WW * 0.5f) / (float)WW;
    g[3] = (my - (float)HH * 0.5f) / (float)HH;
    float xc[4];
#pragma unroll
    for (int i = 0; i < 4; ++i) xc[i] = g[i] - c_mean[i];
    float maha = 0.0f;
#pragma unroll
    for (int i = 0; i < 4; ++i) {
      float acc = 0.0f;
#pragma unroll
      for (int j = 0; j < 4; ++j) acc += c_covinv[i * 4 + j] * xc[j];
      maha += xc[i] * acc;
    }
    float shape = expf(-0.5f * maha);
    float sal = contrast[m] * shape;
    s_mask[m] = maskf;
    Bm[m * 16 + 0] = sal * maskf;
#pragma unroll
    for (int c = 1; c < 16; ++c) Bm[m * 16 + c] = 0.0f;
  }
  __syncthreads();

  // -------- pass 2: E @ (sal*mask) --------
  for (int t = 0; t < 2; ++t) {
    int m0 = (wave * 2 + t) * 16;
    v8f c = {};
    for (int k0 = 0; k0 < NBIN; k0 += 4) {
      int kb = k0 + laneHi * 2;
      v2f a, b;
      a.x = Emat[(size_t)(m0 + laneM) * NBIN + kb];
      a.y = Emat[(size_t)(m0 + laneM) * NBIN + kb + 1];
      b.x = Bm[kb * 16 + laneM];
      b.y = Bm[(kb + 1) * 16 + laneM];
      c = __builtin_amdgcn_wmma_f32_16x16x4_f32(false, a, false, b, (short)0, c, false, false);
    }
    if (laneM == 0) {
#pragma unroll
      for (int r = 0; r < 8; ++r) s_num[m0 + r + laneHi * 8] = c[r];
    }
  }
  __syncthreads();

  // -------- normalize over occupied bins --------
  {
    int m = tid;
    float sn = fmaxf(outA[m * 8 + 5], 1e-8f);  // E @ maskf
    float sm = s_num[m] / sn;
    outA[m * 8 + 6] = sm;
    bool occ = s_mask[m] > 0.0f;
    rmin[m] = occ ? sm : INFINITY;
    rmax[m] = occ ? sm : -INFINITY;
  }
  __syncthreads();
  for (int s = 256; s > 0; s >>= 1) {
    if (tid < s) {
      rmin[tid] = fminf(rmin[tid], rmin[tid + s]);
      rmax[tid] = fmaxf(rmax[tid], rmax[tid + s]);
    }
    __syncthreads();
  }
  {
    int m = tid;
    float mn = rmin[0], mxv = rmax[0];
    float sm = outA[m * 8 + 6];
    float ns = (mxv > mn) ? 255.0f * (sm - mn) / fmaxf(mxv - mn, 1e-30f) : 0.0f;
    normSal[m] = ns;
  }
}

// ---------------- K5: fused gather + horizontal 10-tap sum ----------------
__global__ __launch_bounds__(256) void hblur_kernel(const unsigned short* __restrict__ lin16,
                                                    const float* __restrict__ normSal,
                                                    float* __restrict__ tmp) {
  __shared__ float sal[NBIN];
  __shared__ float row[272];
  int tid = threadIdx.x;
  sal[tid] = normSal[tid];
  sal[tid + 256] = normSal[tid + 256];
  __syncthreads();
  int y = blockIdx.y;
  int x0 = blockIdx.x * 256;
  for (int j = tid; j < 265; j += 256) {
    int x = refl101(x0 - 5 + j, WW);
    row[j] = sal[lin16[(size_t)y * WW + x]];
  }
  __syncthreads();
  float s = 0.0f;
#pragma unroll
  for (int j = 0; j < 10; ++j) s += row[tid + j];
  tmp[(size_t)y * WW + x0 + tid] = s;
}

// ---------------- K6: vertical 10-tap sum + final scale ----------------
__global__ __launch_bounds__(256) void vblur_kernel(const float* __restrict__ tmp,
                                                    float* __restrict__ out) {
  int x = blockIdx.x * 256 + threadIdx.x;
  int y = blockIdx.y;
  float s = 0.0f;
#pragma unroll
  for (int dy = -5; dy <= 4; ++dy) {
    int yy = refl101(y + dy, HH);
    s += tmp[(size_t)yy * WW + x];
  }
  // (sum / 100) / 255
  out[(size_t)y * WW + x] = s * (1.0f / 25500.0f);
}

extern "C" void kernel_launch(void* const* d_in, const int* in_sizes, int n_in,
                              void* d_out, int out_size, void* d_ws, size_t ws_size,
                              hipStream_t stream) {
  (void)in_sizes; (void)n_in; (void)out_size; (void)ws_size;
  const float* img = (const float*)d_in[0];
  float* out = (float*)d_out;
  char* ws = (char*)d_ws;

  int* mm = (int*)(ws + OFF_MM);
  float* binstats = (float*)(ws + OFF_BIN);
  float* contrast = (float*)(ws + OFF_CONTRAST);
  float* normSal = (float*)(ws + OFF_NORMSAL);
  float* Emat = (float*)(ws + OFF_EMAT);
  float* partials = (float*)(ws + OFF_PART);
  unsigned short* lin16 = (unsigned short*)(ws + OFF_LIN);
  float* tmp = (float*)(ws + OFF_TMP);

  init_kernel<<<dim3(1), dim3(64), 0, stream>>>(mm);
  minmax_kernel<<<dim3(2048), dim3(256), 0, stream>>>(img, mm);
  bin_kernel<<<dim3(NB2), dim3(256), 0, stream>>>(img, mm, partials, lin16);
  reduce_bins<<<dim3(10), dim3(256), 0, stream>>>(partials, binstats);
  emat_kernel<<<dim3(NBIN), dim3(256), 0, stream>>>(mm, binstats, Emat, contrast);
  bilateral_wmma_kernel<<<dim3(1), dim3(512), 0, stream>>>(Emat, binstats, contrast, normSal);
  hblur_kernel<<<dim3(WW / 256, HH), dim3(256), 0, stream>>>(lin16, normSal, tmp);
  vblur_kernel<<<dim3(WW / 256, HH), dim3(256), 0, stream>>>(tmp, out);
}